// NeuralMemory_65575560675771
// MI455X (gfx1250) — compile-verified
//
#include <hip/hip_runtime.h>
#include <math.h>

// Problem sizes (fixed by the reference)
#define Bz 4
#define Lz 512
#define Dz 128
#define Hz 256
#define Nz (Bz * Lz) // 2048 rows of x flattened

typedef __attribute__((ext_vector_type(2))) float v2f;
typedef __attribute__((ext_vector_type(8))) float v8f;

// ---------- helpers ----------
__device__ __forceinline__ unsigned short f2bf(float f) {
  union { float f; unsigned int u; } v; v.f = f;
  unsigned int u = v.u;
  unsigned int r = u + 0x7FFFu + ((u >> 16) & 1u); // round-to-nearest-even
  return (unsigned short)(r >> 16);
}
__device__ __forceinline__ float bf2f(unsigned short h) {
  union { unsigned int u; float f; } v; v.u = ((unsigned int)h) << 16;
  return v.f;
}
// jax.nn.gelu default is the tanh approximation
__device__ __forceinline__ float gelu_f(float x) {
  const float c = 0.7978845608028654f, a = 0.044715f;
  float u = c * (x + a * x * x * x);
  return 0.5f * x * (1.f + tanhf(u));
}
__device__ __forceinline__ float gelu_grad(float x) {
  const float c = 0.7978845608028654f, a = 0.044715f;
  float x2 = x * x;
  float u = c * (x + a * x2 * x);
  float th = tanhf(u);
  float sech2 = 1.f - th * th;
  return 0.5f * (1.f + th) + 0.5f * x * sech2 * c * (1.f + 3.f * a * x2);
}

// ---------- kernel 1: k/v projections via V_WMMA_F32_16X16X4_F32 ----------
// Y[2048,128] = X[2048,128] @ W[128,128] + b, for Wk and Wv.
// One wave per 16x16 output tile; K=128 in 32 WMMA steps of 4.
__global__ __launch_bounds__(256) void nm_proj_wmma(
    const float* __restrict__ x,
    const float* __restrict__ Wk_w, const float* __restrict__ Wk_b,
    const float* __restrict__ Wv_w, const float* __restrict__ Wv_b,
    float* __restrict__ k_all, float* __restrict__ v_all) {
  int lane = threadIdx.x & 31;
  int wave = threadIdx.x >> 5;
  int job  = blockIdx.x * 8 + wave;   // 2048 jobs total
  int mat  = job >> 10;               // 0 -> k, 1 -> v  (1024 tiles each)
  int tile = job & 1023;
  int mt = tile >> 3, nt = tile & 7;  // 128 M-tiles x 8 N-tiles
  const float* W    = mat ? Wv_w : Wk_w;
  const float* bias = mat ? Wv_b : Wk_b;
  float* Y          = mat ? v_all : k_all;

  int m0 = mt * 16, n0 = nt * 16;
  int r  = lane & 15;        // row (A) / col (B,C,D) within tile
  int kh = (lane >> 4) << 1; // lanes 16-31 carry K+2,K+3 (ISA 16x4 f32 layout)

  v8f acc = {};
  for (int kb = 0; kb < Dz; kb += 4) {
    v2f a, b;
    const float* xr = x + (size_t)(m0 + r) * Dz + kb + kh;
    a.x = xr[0];
    a.y = xr[1];
    b.x = W[(size_t)(kb + kh + 0) * Dz + n0 + r];
    b.y = W[(size_t)(kb + kh + 1) * Dz + n0 + r];
    acc = __builtin_amdgcn_wmma_f32_16x16x4_f32(
        false, a, false, b, (short)0, acc, false, false);
  }
  // C/D layout: VGPR v holds row m0+v (lanes 0-15) and m0+v+8 (lanes 16-31)
  int half = lane >> 4;
  int n = n0 + r;
  float bn = bias[n];
#pragma unroll
  for (int vi = 0; vi < 8; ++vi) {
    int m = m0 + vi + half * 8;
    Y[(size_t)m * Dz + n] = acc[vi] + bn;
  }
}

// ---------- kernel 2: hyper projection -> eta/alpha/theta ----------
__global__ __launch_bounds__(256) void nm_hyper(
    const float* __restrict__ x, const float* __restrict__ hyper_w,
    const float* __restrict__ hyper_b, float* __restrict__ scal) {
  int nidx = blockIdx.x * 256 + threadIdx.x;
  if (nidx >= Nz) return;
  const float* xr = x + (size_t)nidx * Dz;
  float h0 = hyper_b[0], h1 = hyper_b[1], h2 = hyper_b[2];
  for (int d = 0; d < Dz; ++d) {
    float xv = xr[d];
    h0 += xv * hyper_w[d * 3 + 0];
    h1 += xv * hyper_w[d * 3 + 1];
    h2 += xv * hyper_w[d * 3 + 2];
  }
  float eta   = 1.f / (1.f + expf(-h0));
  float alpha = 1.f / (1.f + expf(-h1));
  float sp    = (h2 > 20.f) ? h2 : log1pf(expf(h2));
  scal[nidx * 3 + 0] = eta;
  scal[nidx * 3 + 1] = alpha;
  scal[nidx * 3 + 2] = sp * 0.01f + 1e-6f;
}

// ---------- kernel 3: sequential fast-weight scan ----------
// One workgroup (512 threads, 16 wave32) per batch. W1/M1/W2/M2 live in LDS as
// bf16 (4 x 64KB = 256KB <= 320KB/WGP); all arithmetic in f32.
__global__ __launch_bounds__(512) void nm_scan(
    const float* __restrict__ k_all, const float* __restrict__ v_all,
    const float* __restrict__ scal,
    const float* __restrict__ mem_W1, const float* __restrict__ mem_W2,
    float* __restrict__ y) {
  extern __shared__ char smem[];
  unsigned short* W1h = (unsigned short*)smem;   // [Dz*Hz], idx d*Hz+h
  unsigned short* M1h = W1h + Dz * Hz;
  unsigned short* W2h = M1h + Dz * Hz;           // [Hz*Dz], idx h*Dz+d
  unsigned short* M2h = W2h + Hz * Dz;
  float* sk   = (float*)(M2h + Hz * Dz); // [Dz]
  float* sv   = sk + Dz;                 // [Dz]
  float* hpre = sv + Dz;                 // [Hz]
  float* hact = hpre + Hz;               // [Hz]
  float* se   = hact + Hz;               // [Dz]
  float* dhp  = se + Dz;                 // [Hz]
  float* red  = dhp + Hz;                // [512]

  const int tid = threadIdx.x;
  const int b   = blockIdx.x;

  // init: W = broadcast(learned init), M = 0
  for (int i = tid; i < Dz * Hz; i += 512) {
    W1h[i] = f2bf(mem_W1[i]);
    M1h[i] = 0;
    W2h[i] = f2bf(mem_W2[i]);
    M2h[i] = 0;
  }
  __syncthreads();

  for (int t = 0; t < Lz; ++t) {
    const int n = b * Lz + t;
    if (tid < Dz) {
      sk[tid] = k_all[(size_t)n * Dz + tid];
      sv[tid] = v_all[(size_t)n * Dz + tid];
      if (t + 1 < Lz) {
        __builtin_prefetch(&k_all[(size_t)(n + 1) * Dz + tid], 0, 1);
        __builtin_prefetch(&v_all[(size_t)(n + 1) * Dz + tid], 0, 1);
      }
    }
    const float eta   = scal[n * 3 + 0];
    const float alpha = scal[n * 3 + 1];
    const float theta = scal[n * 3 + 2];
    const float oma   = 1.f - alpha;
    __syncthreads();

    // stage 1: hpre[h] = sum_d k[d] * W1[d,h]   (2 threads per h)
    {
      int h = tid >> 1, hf = tid & 1;
      float s = 0.f;
      int d0 = hf * 64;
#pragma unroll 8
      for (int j = 0; j < 64; ++j) {
        int d = d0 + j;
        s += sk[d] * bf2f(W1h[d * Hz + h]);
      }
      red[tid] = s;
    }
    __syncthreads();
    if (tid < Hz) {
      float p = red[2 * tid] + red[2 * tid + 1];
      hpre[tid] = p;
      hact[tid] = gelu_f(p);
    }
    __syncthreads();

    // stage 2: pred[d] = sum_h hact[h] * W2[h,d]; y = pred; e = pred - v
    {
      int d = tid >> 2, q = tid & 3;
      float s = 0.f;
      int h0 = q * 64;
#pragma unroll 8
      for (int j = 0; j < 64; ++j) {
        int h = h0 + j;
        s += hact[h] * bf2f(W2h[h * Dz + d]);
      }
      red[tid] = s;
    }
    __syncthreads();
    if (tid < Dz) {
      float pred = red[4 * tid] + red[4 * tid + 1] + red[4 * tid + 2] + red[4 * tid + 3];
      y[(size_t)n * Dz + tid] = pred; // v_hat uses pre-update memory
      se[tid] = pred - sv[tid];
    }
    __syncthreads();

    // stage 3 (fused): dh[h] partial = sum_d e[d]*W2_old[h,d], and update
    // W2/M2 in the same pass (each element owned by exactly one thread).
    {
      int h = tid >> 1, hf = tid & 1;
      float ha = hact[h];
      float acc = 0.f;
      int d0 = hf * 64;
#pragma unroll 4
      for (int j = 0; j < 64; ++j) {
        int d = d0 + j;
        int idx = h * Dz + d;
        float w2 = bf2f(W2h[idx]);
        float m2 = bf2f(M2h[idx]);
        float ev = se[d];
        acc += ev * w2;                       // uses OLD W2
        float m2n = eta * m2 - theta * (ha * ev); // g2 = hact x e (rank-1)
        float w2n = oma * w2 + m2n;
        M2h[idx] = f2bf(m2n);
        W2h[idx] = f2bf(w2n);
      }
      red[tid] = acc;
    }
    __syncthreads();
    if (tid < Hz) {
      float dh = red[2 * tid] + red[2 * tid + 1];
      dhp[tid] = dh * gelu_grad(hpre[tid]);
    }
    __syncthreads();

    // stage 4: W1/M1 update, g1 = k x dhpre (rank-1)
    {
      int d = tid >> 2, q = tid & 3;
      float kd = sk[d];
      int h0 = q * 64;
#pragma unroll 4
      for (int j = 0; j < 64; ++j) {
        int h = h0 + j;
        int idx = d * Hz + h;
        float w1 = bf2f(W1h[idx]);
        float m1 = bf2f(M1h[idx]);
        float m1n = eta * m1 - theta * (kd * dhp[h]);
        float w1n = oma * w1 + m1n;
        M1h[idx] = f2bf(m1n);
        W1h[idx] = f2bf(w1n);
      }
    }
    __syncthreads();
  }
}

// ---------- host launcher ----------
extern "C" void kernel_launch(void* const* d_in, const int* in_sizes, int n_in,
                              void* d_out, int out_size, void* d_ws, size_t ws_size,
                              hipStream_t stream) {
  const float* x       = (const float*)d_in[0];
  const float* Wk_w    = (const float*)d_in[1];
  const float* Wk_b    = (const float*)d_in[2];
  const float* Wv_w    = (const float*)d_in[3];
  const float* Wv_b    = (const float*)d_in[4];
  const float* hyper_w = (const float*)d_in[5];
  const float* hyper_b = (const float*)d_in[6];
  const float* mem_W1  = (const float*)d_in[7];
  const float* mem_W2  = (const float*)d_in[8];

  float* ws    = (float*)d_ws;
  float* k_all = ws;                         // [Nz*Dz]
  float* v_all = k_all + (size_t)Nz * Dz;    // [Nz*Dz]
  float* scal  = v_all + (size_t)Nz * Dz;    // [Nz*3]

  nm_proj_wmma<<<256, 256, 0, stream>>>(x, Wk_w, Wk_b, Wv_w, Wv_b, k_all, v_all);
  nm_hyper<<<Nz / 256, 256, 0, stream>>>(x, hyper_w, hyper_b, scal);

  size_t lds = (size_t)4 * Dz * Hz * sizeof(unsigned short)   // bf16 state: 256 KB
             + (size_t)(Dz + Dz + Hz + Hz + Dz + Hz + 512) * sizeof(float);
  nm_scan<<<Bz, 512, lds, stream>>>(k_all, v_all, scal, mem_W1, mem_W2, (float*)d_out);
}